// new_loss_fn_46265387712705
// MI455X (gfx1250) — compile-verified
//
#include <hip/hip_runtime.h>
#include <hip/hip_bf16.h>

typedef __attribute__((ext_vector_type(2))) float v2f;
typedef __attribute__((ext_vector_type(8))) float v8f;

#define PSTRIDE 144   // 129 live floats per block partial, padded to 9*16 for WMMA tiles
#define WPB 8         // waves per block (wave32)
#define TPB 256
#define MAXB 1024

// ---------------------------------------------------------------------------
// Kernel 1: single streaming pass over predictions [N x 64].
// Wave handles one row per iteration: lane l loads float2 = cols (2l, 2l+1),
// accumulating exp-sums + target counts for its two fixed columns and the
// picked log-prob.  Deterministic LDS reduction -> per-block partials in ws.
// ---------------------------------------------------------------------------
__global__ __launch_bounds__(TPB) void nl_main(const float* __restrict__ pred,
                                               const int* __restrict__ tgt,
                                               float* __restrict__ partials,
                                               int n, int nblocks) {
  const int lane = threadIdx.x & 31;
  const int wave = threadIdx.x >> 5;
  const int col0 = lane * 2;
  const int col1 = col0 + 1;
  const int W = nblocks * WPB;          // total waves in grid

  float s0 = 0.f, s1 = 0.f;             // exp column sums (cols col0, col1)
  float c0 = 0.f, c1 = 0.f;             // class counts   (cols col0, col1)
  float pick = 0.f;                     // sum of picked log-probs

  for (int row = blockIdx.x * WPB + wave; row < n; row += W) {
    const float2 v = *reinterpret_cast<const float2*>(pred + (size_t)row * 64 + col0);
    const int t = tgt[row];
    if (row + W < n)  // gfx1250: lowers to global_prefetch_b8
      __builtin_prefetch(pred + (size_t)(row + W) * 64 + col0, 0, 1);
    s0 += __expf(v.x);
    s1 += __expf(v.y);
    if (t == col0)      { c0 += 1.f; pick += v.x; }
    else if (t == col1) { c1 += 1.f; pick += v.y; }
  }

  __shared__ float lsum[WPB * 64];
  __shared__ float lcnt[WPB * 64];
  __shared__ float lpick[TPB];
  lsum[wave * 64 + col0] = s0;
  lsum[wave * 64 + col1] = s1;
  lcnt[wave * 64 + col0] = c0;
  lcnt[wave * 64 + col1] = c1;
  lpick[threadIdx.x] = pick;
  __syncthreads();

  float* out = partials + (size_t)blockIdx.x * PSTRIDE;
  const int tid = threadIdx.x;
  if (tid < 64) {
    float a = 0.f, b = 0.f;
#pragma unroll
    for (int w = 0; w < WPB; ++w) {
      a += lsum[w * 64 + tid];
      b += lcnt[w * 64 + tid];
    }
    out[tid]      = a;   // colsum partial, q = 0..63
    out[64 + tid] = b;   // count partial,  q = 64..127
  } else if (tid == 64) {
    float p = 0.f;
    for (int i = 0; i < TPB; ++i) p += lpick[i];
    out[128] = p;        // picked partial, q = 128
  } else if (tid > 64 && tid < 64 + (PSTRIDE - 128)) {
    out[64 + tid] = 0.f; // zero pad q = 129..143 so WMMA tiles are clean
  }
}

// ---------------------------------------------------------------------------
// Kernel 2: cross-block reduction tot[q] = sum_b P[b][q] done as a matrix
// product ones[16x4] x P-chunk[4x16] with V_WMMA_F32_16X16X4_F32 (exact f32).
// With A == all-ones, D[m][n] = sum_k B[k][n] + C[m][n]  (K-order invariant).
// Single full wave => EXEC all-ones at every WMMA, uniform loop bounds.
// ---------------------------------------------------------------------------
__global__ __launch_bounds__(32) void nl_reduce(const float* __restrict__ partials,
                                                float* __restrict__ result,
                                                int n, int nblocks) {
  const int lane = threadIdx.x;             // 0..31, one wave
  __shared__ float tot[PSTRIDE];

  const v2f aones = {1.0f, 1.0f};           // 16x4 ones matrix (same in all lanes)
  const int qlo = lane & 15;                // N index within tile
  const int bo  = (lane >> 4) << 1;         // half-wave K offset (0 or 2)

  for (int tile = 0; tile < PSTRIDE / 16; ++tile) {
    v8f acc = {};
    const int q = tile * 16 + qlo;
    for (int b = 0; b < nblocks; b += 4) {  // nblocks is a multiple of 4
      const float* p = partials + (size_t)(b + bo) * PSTRIDE + q;
      v2f bv;
      bv.x = p[0];                          // row b+bo,   column q
      bv.y = p[PSTRIDE];                    // row b+bo+1, column q
      acc = __builtin_amdgcn_wmma_f32_16x16x4_f32(
          /*neg_a=*/false, aones, /*neg_b=*/false, bv,
          /*c_mod=*/(short)0, acc, /*reuse_a=*/false, /*reuse_b=*/false);
    }
    // D row 0: VGPR0 on lanes 0..15 (M=0, N=lane)
    if (lane < 16) tot[tile * 16 + lane] = acc[0];
  }
  __syncthreads();

  if (lane == 0) {
    float sq = 0.f;
    for (int c = 0; c < 64; ++c) {
      const float d = tot[64 + c] - tot[c]; // class_dist - col_sum
      sq += d * d;
    }
    // nll + penalty = (-sum picked + sum d^2) / n
    result[0] = (sq - tot[128]) / (float)n;
  }
}

// ---------------------------------------------------------------------------
extern "C" void kernel_launch(void* const* d_in, const int* in_sizes, int n_in,
                              void* d_out, int out_size, void* d_ws, size_t ws_size,
                              hipStream_t stream) {
  const float* pred = (const float*)d_in[0];   // [N x 64] fp32 log-probs
  const int*   tgt  = (const int*)d_in[1];     // [N] int32 class ids
  const int    n    = in_sizes[1];             // N = number of rows

  float* partials = (float*)d_ws;
  int nb = (int)(ws_size / (PSTRIDE * sizeof(float)));
  if (nb > MAXB) nb = MAXB;
  nb &= ~3;                                    // multiple of 4 for K=4 WMMA
  if (nb < 4) nb = 4;

  nl_main<<<nb, TPB, 0, stream>>>(pred, tgt, partials, n, nb);
  nl_reduce<<<1, 32, 0, stream>>>(partials, (float*)d_out, n, nb);
}